// GCN_Align_20023137534370
// MI455X (gfx1250) — compile-verified
//
#include <hip/hip_runtime.h>
#include <hip/hip_bf16.h>

typedef float v2f __attribute__((ext_vector_type(2)));
typedef float v8f __attribute__((ext_vector_type(8)));

#define TM 64
#define TN 128
#define TK 16

#define WMMA_F32(a, b, c) \
    __builtin_amdgcn_wmma_f32_16x16x4_f32(false, (a), false, (b), (short)0, (c), false, false)

// Async memory->LDS copy (CDNA5 GLOBAL_LOAD_ASYNC_TO_LDS_B128, ASYNCcnt-tracked).
// ldsOff: LDS byte offset = low 32 bits of the flat shared-memory pointer.
__device__ __forceinline__ void async_copy_b128(unsigned ldsOff, const float* gptr) {
    asm volatile("global_load_async_to_lds_b128 %0, %1, off"
                 :: "v"(ldsOff), "v"(gptr)
                 : "memory");
}
// ASYNCcnt completes in-order: waiting <=3 after issuing 3 new copies
// guarantees the previous tile's 3 copies have landed in LDS.
__device__ __forceinline__ void wait_asynccnt0() {
    asm volatile("s_wait_asynccnt 0x0" ::: "memory");
}
__device__ __forceinline__ void wait_asynccnt3() {
    asm volatile("s_wait_asynccnt 0x3" ::: "memory");
}

// C[M,N] = A[M,K] x B[K,N], row-major fp32, via V_WMMA_F32_16X16X4_F32.
// 256 threads = 8 waves in a 2(M) x 4(N) wave grid; each wave owns a 32x32
// register tile (2 A-frags x 2 B-frags outer product, 4 accumulators).
// Double-buffered LDS: tile k+1 is async-copied while tile k is consumed.
// OOB policy: only K-overrun must be zero-filled; M-row / N-col overruns are
// clamped to valid memory and masked at the store.
__global__ __launch_bounds__(256)
void gemm_f32_wmma(const float* __restrict__ A, const float* __restrict__ B,
                   float* __restrict__ C, int M, int N, int K) {
    __shared__ float As[2][TM][TK + 4];   // row stride 20 floats (16B-aligned, conflict-free)
    __shared__ float Bs[2][TK][TN + 4];   // row stride 132 floats

    const int t    = threadIdx.x;
    const int wave = t >> 5;
    const int lane = t & 31;
    const int wm   = wave & 1;
    const int wn   = wave >> 1;
    const int blockM = blockIdx.x * TM;
    const int blockN = blockIdx.y * TN;

    v8f acc00 = {}, acc01 = {}, acc10 = {}, acc11 = {};

    // Cooperative fill indices: A = one b128/thread, B = two b128/thread.
    const int arow  = t >> 2, acol = (t & 3) << 2;     // As: 64 x 16
    const int aRowG = min(blockM + arow, M - 1);       // clamp (masked at store)
    const int br0   = t >> 5, bc0 = (t & 31) << 2;     // Bs: 16 x 128, two slots
    const int br1   = br0 + 8;
    const int gcB0  = min(blockN + bc0, N - 4);        // clamp (masked at store)

    const unsigned asO[2]  = { (unsigned)(uintptr_t)&As[0][arow][acol],
                               (unsigned)(uintptr_t)&As[1][arow][acol] };
    const unsigned bsO0[2] = { (unsigned)(uintptr_t)&Bs[0][br0][bc0],
                               (unsigned)(uintptr_t)&Bs[1][br0][bc0] };
    const unsigned bsO1[2] = { (unsigned)(uintptr_t)&Bs[0][br1][bc0],
                               (unsigned)(uintptr_t)&Bs[1][br1][bc0] };

    // Running global pointers (strength-reduced: +TK for A, +TK*N for B).
    const size_t bStride = (size_t)TK * N;
    const float* aN  = A + (size_t)aRowG * K + acol;
    const float* bN0 = B + (size_t)br0 * N + gcB0;
    const float* bN1 = B + (size_t)br1 * N + gcB0;

    // Synchronous zero-padded fill for the single partial K-tile.
    auto syncFill = [&](int bi, int k0) {
        #pragma unroll
        for (int j = 0; j < 4; ++j) {
            const int gk = k0 + acol + j;
            As[bi][arow][acol + j] = (gk < K) ? A[(size_t)aRowG * K + gk] : 0.0f;
        }
        #pragma unroll
        for (int p = 0; p < 2; ++p) {
            const int brp = p ? br1 : br0;
            const int gk  = k0 + brp;
            float4 bv = make_float4(0.f, 0.f, 0.f, 0.f);
            if (gk < K) bv = *(const float4*)(B + (size_t)gk * N + gcB0);
            *(float4*)&Bs[bi][brp][bc0] = bv;
        }
    };

    // ---- prologue: stage tile 0 into buffer 0 ----
    bool curAsync;
    if (TK <= K) {
        async_copy_b128(asO[0],  aN);
        async_copy_b128(bsO0[0], bN0);
        async_copy_b128(bsO1[0], bN1);
        curAsync = true;
    } else {
        syncFill(0, 0);
        curAsync = false;
    }
    aN += TK;  bN0 += bStride;  bN1 += bStride;

    // Fragment read indices (ISA 32-bit A 16x4 layout: lanes 0-15 -> K=kk+0,1;
    // lanes 16-31 -> K=kk+2,3).
    const int half = (lane < 16) ? 0 : 2;
    const int lidx = lane & 15;
    const int mA0  = wm * 32 + lidx;
    const int mA1  = mA0 + 16;
    const int nB0  = wn * 32 + lidx;
    const int nB1  = nB0 + 16;

    int buf = 0;
    for (int k0 = 0; k0 < K; k0 += TK) {
        // ---- stage tile k0+TK into the other buffer while tile k0 is in flight ----
        const int kNext = k0 + TK;
        bool nextAsync = false;
        if (kNext < K) {
            if (kNext + TK <= K) {
                async_copy_b128(asO[buf ^ 1],  aN);
                async_copy_b128(bsO0[buf ^ 1], bN0);
                async_copy_b128(bsO1[buf ^ 1], bN1);
                nextAsync = true;
                if (kNext + 2 * TK <= K) {            // nudge tile k0+2*TK toward L2
                    __builtin_prefetch(aN + TK, 0, 3);
                    __builtin_prefetch(bN0 + bStride, 0, 3);
                }
            } else {
                syncFill(buf ^ 1, kNext);
            }
            aN += TK;  bN0 += bStride;  bN1 += bStride;
        }
        // Tile k0 must be resident; the newest 3 copies (tile k0+TK) may remain.
        if (curAsync) { if (nextAsync) wait_asynccnt3(); else wait_asynccnt0(); }
        __syncthreads();

        #pragma unroll
        for (int kk = 0; kk < TK; kk += 4) {
            v2f a0, a1, b0, b1;
            a0.x = As[buf][mA0][kk + half];  a0.y = As[buf][mA0][kk + half + 1];
            a1.x = As[buf][mA1][kk + half];  a1.y = As[buf][mA1][kk + half + 1];
            b0.x = Bs[buf][kk + half][nB0];  b0.y = Bs[buf][kk + half + 1][nB0];
            b1.x = Bs[buf][kk + half][nB1];  b1.y = Bs[buf][kk + half + 1][nB1];
            acc00 = WMMA_F32(a0, b0, acc00);
            acc01 = WMMA_F32(a0, b1, acc01);
            acc10 = WMMA_F32(a1, b0, acc10);
            acc11 = WMMA_F32(a1, b1, acc11);
        }
        __syncthreads();   // all waves done reading buf before it is refilled next iter
        buf ^= 1;
        curAsync = nextAsync;
    }

    // C/D layout: VGPR i -> M = i (lanes 0-15) / M = 8+i (lanes 16-31), N = lane&15.
    const int mOff = (lane < 16) ? 0 : 8;
    #pragma unroll
    for (int i = 0; i < 8; ++i) {
        const int gr0 = blockM + wm * 32 + mOff + i;
        const int gr1 = gr0 + 16;
        const int gc0 = blockN + wn * 32 + lidx;
        const int gc1 = gc0 + 16;
        if (gr0 < M) {
            if (gc0 < N) C[(size_t)gr0 * N + gc0] = acc00[i];
            if (gc1 < N) C[(size_t)gr0 * N + gc1] = acc01[i];
        }
        if (gr1 < M) {
            if (gc0 < N) C[(size_t)gr1 * N + gc0] = acc10[i];
            if (gc1 < N) C[(size_t)gr1 * N + gc1] = acc11[i];
        }
    }
}

// Y[row[e], :] += vals[e] * X[col[e], :]   (Y pre-zeroed). One wave32 per edge.
__global__ __launch_bounds__(256)
void spmm_scatter(const int* __restrict__ row, const int* __restrict__ col,
                  const float* __restrict__ vals, const float* __restrict__ X,
                  float* __restrict__ Y, int E, int dim) {
    const int e    = blockIdx.x * 8 + (threadIdx.x >> 5);
    const int lane = threadIdx.x & 31;
    if (e >= E) return;
    const int   r = row[e];
    const int   c = col[e];
    const float v = vals[e];
    const float* __restrict__ src = X + (size_t)c * dim;
    float* __restrict__ dst       = Y + (size_t)r * dim;
    for (int j = lane; j < dim; j += 32)
        atomicAdd(&dst[j], v * src[j]);
}

// n4 = element count / 4 (all row sizes here are multiples of 4).
__global__ __launch_bounds__(256)
void fill_zero4(float4* __restrict__ p, int n4) {
    const int i = blockIdx.x * 256 + threadIdx.x;
    if (i < n4) p[i] = make_float4(0.f, 0.f, 0.f, 0.f);
}

__global__ __launch_bounds__(256)
void relu_inplace4(float4* __restrict__ p, int n4) {
    const int i = blockIdx.x * 256 + threadIdx.x;
    if (i < n4) {
        float4 v = p[i];
        v.x = fmaxf(v.x, 0.f); v.y = fmaxf(v.y, 0.f);
        v.z = fmaxf(v.z, 0.f); v.w = fmaxf(v.w, 0.f);
        p[i] = v;
    }
}

// x = relu(x); x /= max(||x||_2, 1e-12) per row. One wave32 per row.
__global__ __launch_bounds__(256)
void relu_l2norm_rows(float* __restrict__ X, int nrows, int dim) {
    const int r    = blockIdx.x * 8 + (threadIdx.x >> 5);
    const int lane = threadIdx.x & 31;
    if (r >= nrows) return;
    float* __restrict__ rowp = X + (size_t)r * dim;
    float ss = 0.0f;
    for (int j = lane; j < dim; j += 32) {
        const float v = fmaxf(rowp[j], 0.0f);
        ss += v * v;
    }
    #pragma unroll
    for (int off = 16; off > 0; off >>= 1)
        ss += __shfl_xor(ss, off, 32);
    const float scale = 1.0f / fmaxf(sqrtf(ss), 1e-12f);
    for (int j = lane; j < dim; j += 32)
        rowp[j] = fmaxf(rowp[j], 0.0f) * scale;
}

static inline void launch_gemm(const float* A, const float* B, float* C,
                               int M, int N, int K, hipStream_t s) {
    dim3 grid((M + TM - 1) / TM, (N + TN - 1) / TN);
    gemm_f32_wmma<<<grid, 256, 0, s>>>(A, B, C, M, N, K);
}

static inline void launch_zero(float* p, int n, hipStream_t s) {
    const int n4 = n >> 2;
    fill_zero4<<<(n4 + 255) / 256, 256, 0, s>>>((float4*)p, n4);
}

static inline void launch_relu(float* p, int n, hipStream_t s) {
    const int n4 = n >> 2;
    relu_inplace4<<<(n4 + 255) / 256, 256, 0, s>>>((float4*)p, n4);
}

extern "C" void kernel_launch(void* const* d_in, const int* in_sizes, int n_in,
                              void* d_out, int out_size, void* d_ws, size_t ws_size,
                              hipStream_t stream) {
    const int   D  = 1000;                 // structural / attribute input width
    const int   Da = 100;                  // attribute hidden width
    const int   E  = in_sizes[4];          // edges per graph
    const int   N  = in_sizes[0] / D;      // nodes per graph

    const float* emb[2]   = { (const float*)d_in[0], (const float*)d_in[1] };
    const float* attr[2]  = { (const float*)d_in[2], (const float*)d_in[3] };
    const int*   rowi[2]  = { (const int*)d_in[4],   (const int*)d_in[7] };
    const int*   coli[2]  = { (const int*)d_in[5],   (const int*)d_in[8] };
    const float* valf[2]  = { (const float*)d_in[6], (const float*)d_in[9] };
    const float* W_s0     = (const float*)d_in[10];
    const float* W_s1     = (const float*)d_in[11];
    const float* W_a1[2]  = { (const float*)d_in[12], (const float*)d_in[13] };
    const float* W_a2     = (const float*)d_in[14];

    float* out   = (float*)d_out;
    float* outStruct[2] = { out, out + (size_t)N * D };
    float* outA0 = out + (size_t)2 * N * D;
    float* outAttr[2]   = { outA0, outA0 + (size_t)N * Da };

    float* ws0 = (float*)d_ws;              // [N, D] scratch
    float* ws1 = ws0 + (size_t)N * D;       // [N, D] scratch

    const int ND  = N * D;
    const int NDa = N * Da;
    const int spmmGrid = (E + 7) / 8;

    for (int g = 0; g < 2; ++g) {
        // ---------- structural branch: 2 GCN layers + l2norm ----------
        launch_gemm(emb[g], W_s0, ws0, N, D, D, stream);
        launch_zero(ws1, ND, stream);
        spmm_scatter<<<spmmGrid, 256, 0, stream>>>(rowi[g], coli[g], valf[g], ws0, ws1, E, D);
        launch_relu(ws1, ND, stream);

        launch_gemm(ws1, W_s1, ws0, N, D, D, stream);
        launch_zero(outStruct[g], ND, stream);
        spmm_scatter<<<spmmGrid, 256, 0, stream>>>(rowi[g], coli[g], valf[g], ws0,
                                                   outStruct[g], E, D);
        relu_l2norm_rows<<<(N + 7) / 8, 256, 0, stream>>>(outStruct[g], N, D);

        // ---------- attribute branch: 2 GCN layers ----------
        launch_gemm(attr[g], W_a1[g], ws0, N, Da, D, stream);
        launch_zero(ws1, NDa, stream);
        spmm_scatter<<<spmmGrid, 256, 0, stream>>>(rowi[g], coli[g], valf[g], ws0, ws1, E, Da);
        launch_relu(ws1, NDa, stream);

        launch_gemm(ws1, W_a2, ws0, N, Da, Da, stream);
        launch_zero(outAttr[g], NDa, stream);
        spmm_scatter<<<spmmGrid, 256, 0, stream>>>(rowi[g], coli[g], valf[g], ws0,
                                                   outAttr[g], E, Da);
        launch_relu(outAttr[g], NDa, stream);
    }
}